// Gsaloss_10823317586699
// MI455X (gfx1250) — compile-verified
//
#include <hip/hip_runtime.h>
#include <hip/hip_bf16.h>
#include <stdint.h>

// GSA loss: token[2,4096,4096] f32 streamed once (128 MB, HBM-bound ~5.5us @23.3TB/s).
// Per 16-row stripe: TDM (tensor_load_to_lds) double-buffered 16x512 tiles into LDS,
// reduce with chained v_wmma_f32_16x16x4_f32 against W = [ones, p_j], fixed-order
// deterministic reductions -> 4 scalars -> loss.

#define HW      4096
#define ROWS    16
#define CHUNK   512
#define LSTRIDE 516              // 512 + TDM pad (2 dwords per 256) -> bank-conflict-free
#define THREADS 256
#define NCHUNK  (HW / CHUNK)     // 8
#define STRIPES 512              // 2 batches * (4096/16)

typedef __attribute__((ext_vector_type(2))) float        v2f;
typedef __attribute__((ext_vector_type(8))) float        v8f;
typedef __attribute__((ext_vector_type(4))) unsigned int u32x4;
typedef __attribute__((ext_vector_type(4))) int          i32x4;
typedef __attribute__((ext_vector_type(8))) int          i32x8;

// Issue one TDM load of a 16x512 f32 tile (row stride 4096 elems) -> LDS.
// LDS padding: +2 dwords every 256 dwords => row stride 516 dwords.
__device__ __forceinline__ void tdm_load_tile(unsigned lds_off, uint64_t gaddr)
{
    u32x4 g0;
    g0[0] = 1u;                                        // count=1, user descriptor
    g0[1] = lds_off;                                   // LDS dest (bytes)
    g0[2] = (unsigned)(gaddr & 0xffffffffu);           // global_addr[31:0]
    g0[3] = (unsigned)((gaddr >> 32) & 0x01ffffffu)    // global_addr[56:32]
          | (2u << 30);                                // type=2 (image)

    i32x8 g1;
    g1[0] = (int)((2u << 16)                           // data_size = 4 bytes
          | (1u << 20)                                 // pad_enable
          | (7u << 22)                                 // pad_interval: 256 dwords
          | (1u << 25));                               // pad_amount: 2 dwords
    g1[1] = (int)(4096u << 16);                        // tensor_dim0 = 4096 (lo16)
    g1[2] = (int)(4096u << 16);                        // dim0 hi16=0 | tensor_dim1 lo16 = 4096
    g1[3] = (int)(512u  << 16);                        // dim1 hi16=0 | tile_dim0 = 512
    g1[4] = 16;                                        // tile_dim1 = 16 | tile_dim2 = 0
    g1[5] = 4096;                                      // tensor_dim0_stride lo32
    g1[6] = 0;
    g1[7] = 0;

    i32x4 gz = {};
#if __clang_major__ >= 23
    i32x8 gz8 = {};
    __builtin_amdgcn_tensor_load_to_lds(g0, g1, gz, gz, gz8, 0);
#else
    __builtin_amdgcn_tensor_load_to_lds(g0, g1, gz, gz, 0);
#endif
}

__global__ __launch_bounds__(THREADS)
void gsa_stripe_kernel(const float* __restrict__ pred,
                       const float* __restrict__ token,
                       float* __restrict__ partials)
{
    __shared__ __align__(16) float smem[2][ROWS * LSTRIDE]; // 2 x 33 KB token tiles
    __shared__ __align__(16) float sp[2][CHUNK];            // 2 x 2 KB p chunks
    __shared__ float red[THREADS * 4];                      // 4 KB block reduction

    const int stripe = blockIdx.x;           // 0..511
    const int b      = stripe >> 8;          // 256 stripes per batch
    const int i0     = (stripe & 255) * ROWS;

    const uint64_t tbase = (uint64_t)(uintptr_t)(token + (size_t)b * HW * HW
                                                       + (size_t)i0 * HW);
    const uint64_t pbase = (uint64_t)(uintptr_t)(pred + (size_t)b * HW);
    const float* __restrict__ pb = pred + (size_t)b * HW;

    const int tid    = threadIdx.x;
    const int lane   = tid & 31;
    const int wave   = tid >> 5;             // 0..7
    const int hi     = (lane >> 4) & 1;      // lane half
    const int lane16 = lane & 15;

    v8f acc = {};                            // chained WMMA accumulator (D = 16x16 f32)

    // ---- prologue: load chunk 0 into buffer 0 ----
    if (wave == 0)
        tdm_load_tile((unsigned)(uintptr_t)&smem[0][0], tbase);
    if (tid < 128) {
        const unsigned lds_off = (unsigned)(uintptr_t)&sp[0][tid * 4];
        const unsigned goff    = (unsigned)(tid * 16);
        asm volatile("global_load_async_to_lds_b128 %0, %1, %2"
                     :: "v"(lds_off), "v"(goff), "s"(pbase) : "memory");
    }
    if (wave == 0)
        __builtin_amdgcn_s_wait_tensorcnt(0);
    asm volatile("s_wait_asynccnt 0" ::: "memory");
    __syncthreads();

    for (int c = 0; c < NCHUNK; ++c) {
        const int cur = c & 1;

        // ---- prefetch chunk c+1 into the other buffer (overlaps WMMA below) ----
        if (c + 1 < NCHUNK) {
            const int nxt = cur ^ 1;
            const int j0n = (c + 1) * CHUNK;
            if (wave == 0)
                tdm_load_tile((unsigned)(uintptr_t)&smem[nxt][0],
                              tbase + (uint64_t)j0n * sizeof(float));
            if (tid < 128) {
                const unsigned lds_off = (unsigned)(uintptr_t)&sp[nxt][tid * 4];
                const unsigned goff    = (unsigned)((j0n + tid * 4) * sizeof(float));
                asm volatile("global_load_async_to_lds_b128 %0, %1, %2"
                             :: "v"(lds_off), "v"(goff), "s"(pbase) : "memory");
            }
        }

        // ---- WMMA-reduce current chunk: each wave owns 4 of 32 j-tiles ----
        #pragma unroll
        for (int tt = 0; tt < 4; ++tt) {
            const int jc   = (wave * 4 + tt) * 16;   // tile column base in chunk
            const int bump = (jc >= 256) ? 2 : 0;    // TDM mid-row pad shift
            #pragma unroll
            for (int t = 0; t < 4; ++t) {
                // A fragment (16x4 f32): lane row = lane16, K pair = 2*hi+{0,1}
                const int kb = jc + t * 4 + hi * 2;
                v2f a = *(const v2f*)&smem[cur][lane16 * LSTRIDE + kb + bump];
                // B fragment (4x16): VGPR v -> K = v + 2*hi, N = lane16.
                // W[:,0]=1, W[:,1]=p_j, rest 0.
                const float p0 = sp[cur][jc + t * 4 + 2 * hi + 0];
                const float p1 = sp[cur][jc + t * 4 + 2 * hi + 1];
                v2f bb;
                bb.x = (lane16 == 0) ? 1.0f : ((lane16 == 1) ? p0 : 0.0f);
                bb.y = (lane16 == 0) ? 1.0f : ((lane16 == 1) ? p1 : 0.0f);
                acc = __builtin_amdgcn_wmma_f32_16x16x4_f32(
                          false, a, false, bb, (short)0, acc, false, false);
            }
        }

        // ---- drain prefetch, sync buffers ----
        if (wave == 0)
            __builtin_amdgcn_s_wait_tensorcnt(0);
        asm volatile("s_wait_asynccnt 0" ::: "memory");
        __syncthreads();
    }

    // D layout: VGPR v -> row m = v + 8*hi, col n = lane16.
    // col 0 = row sums r_i, col 1 = sum_j M_ij * p_j.
    float tT = 0.f, tPR = 0.f, tPC = 0.f, tQ = 0.f;
    if (lane16 < 2) {
        #pragma unroll
        for (int v = 0; v < 8; ++v) {
            const int   m   = v + 8 * hi;
            const float pi  = pb[i0 + m];
            const float val = acc[v];
            if (lane16 == 0) { tT  += val; tPR += pi * val; }
            else             { tPC += val; tQ  += pi * val; }
        }
    }
    red[tid * 4 + 0] = tT;
    red[tid * 4 + 1] = tPR;
    red[tid * 4 + 2] = tPC;
    red[tid * 4 + 3] = tQ;
    __syncthreads();

    if (tid < 4) {                            // fixed-order, deterministic
        float s = 0.f;
        for (int i = 0; i < THREADS; ++i) s += red[i * 4 + tid];
        partials[stripe * 4 + tid] = s;
    }
}

__global__ void gsa_final_kernel(const float* __restrict__ partials,
                                 float* __restrict__ out)
{
    if (blockIdx.x == 0 && threadIdx.x == 0) {
        float T = 0.f, PR = 0.f, PC = 0.f, Q = 0.f;
        for (int s = 0; s < STRIPES; ++s) {   // fixed-order, deterministic
            T  += partials[s * 4 + 0];
            PR += partials[s * 4 + 1];
            PC += partials[s * 4 + 2];
            Q  += partials[s * 4 + 3];
        }
        const float s_f  = PR;
        const float s_fa = Q;
        const float s_b  = T - PR;
        const float s_bb = T - PR - PC + Q;
        out[0] = 2.0f - s_fa / s_f - s_bb / s_b;
    }
}

extern "C" void kernel_launch(void* const* d_in, const int* in_sizes, int n_in,
                              void* d_out, int out_size, void* d_ws, size_t ws_size,
                              hipStream_t stream)
{
    const float* pred  = (const float*)d_in[0];   // [2,1,64,64] f32
    const float* token = (const float*)d_in[1];   // [2,4096,4096] f32
    float* partials    = (float*)d_ws;            // needs STRIPES*4*4 = 8 KB scratch
    float* out         = (float*)d_out;           // scalar f32

    gsa_stripe_kernel<<<STRIPES, THREADS, 0, stream>>>(pred, token, partials);
    gsa_final_kernel<<<1, 32, 0, stream>>>(partials, out);
}